// GatedAttention_42674795053308
// MI455X (gfx1250) — compile-verified
//
#include <hip/hip_runtime.h>

typedef __attribute__((ext_vector_type(2))) float v2f;
typedef __attribute__((ext_vector_type(8))) float v8f;

#define BB 8
#define NN 2048
#define HH 512
#define AA 128
#define LDSTRIDE 2052   // 2048 + 4 pad: kills LDS bank conflicts between half-waves

// ---------------------------------------------------------------------------
// Kernel 1: Qw = x@Wq + bq, Kw = x@Wk + bk  (fp32 WMMA 16x16x4, shared A-frags)
// grid = B * N/16 blocks, 256 threads (8 waves); wave w -> column tile w (16 cols)
// ---------------------------------------------------------------------------
__global__ __launch_bounds__(256) void proj_qk(const float* __restrict__ x,
                                               const float* __restrict__ Wq,
                                               const float* __restrict__ bq,
                                               const float* __restrict__ Wk,
                                               const float* __restrict__ bk,
                                               float* __restrict__ Qw,
                                               float* __restrict__ Kw) {
  const int blk  = blockIdx.x;          // b*(N/16) + row tile
  const int b    = blk >> 7;            // N/16 == 128
  const int r0   = (blk & 127) << 4;
  const int wave = threadIdx.x >> 5;
  const int lane = threadIdx.x & 31;
  const int c0   = wave << 4;           // 8 waves * 16 = 128 = A  ✓
  const int lm   = lane & 15;
  const int lh   = lane >> 4;           // 0 or 1: selects K pair {0,1} vs {2,3}

  const float* xrow = x + (size_t)(b * NN + r0 + lm) * HH;

  v8f accq = {};
  v8f acck = {};
  for (int kk = 0; kk < HH / 4; ++kk) {
    const int k = 4 * kk + 2 * lh;
    const v2f a = *(const v2f*)(xrow + k);                 // A frag: x rows
    v2f fq, fk;                                            // B frags: W columns
    fq.x = Wq[(size_t)k * AA + c0 + lm];
    fq.y = Wq[(size_t)(k + 1) * AA + c0 + lm];
    fk.x = Wk[(size_t)k * AA + c0 + lm];
    fk.y = Wk[(size_t)(k + 1) * AA + c0 + lm];
    accq = __builtin_amdgcn_wmma_f32_16x16x4_f32(false, a, false, fq,
                                                 (short)0, accq, false, false);
    acck = __builtin_amdgcn_wmma_f32_16x16x4_f32(false, a, false, fk,
                                                 (short)0, acck, false, false);
  }

  const float biasq = bq[c0 + lm];
  const float biask = bk[c0 + lm];
#pragma unroll
  for (int v = 0; v < 8; ++v) {
    const int row = r0 + v + 8 * lh;        // C layout: VGPR v, halfwave lh
    const size_t idx = (size_t)(b * NN + row) * AA + c0 + lm;
    Qw[idx] = accq[v] + biasq;
    Kw[idx] = acck[v] + biask;
  }
}

// ---------------------------------------------------------------------------
// Kernel 2: g = sigmoid(Qw @ Wv + bv)   (tiny GEMV, one thread per token)
// ---------------------------------------------------------------------------
__global__ void gate_kernel(const float* __restrict__ Qw,
                            const float* __restrict__ Wv,
                            const float* __restrict__ bv,
                            float* __restrict__ g) {
  const int i = blockIdx.x * blockDim.x + threadIdx.x;
  if (i >= BB * NN) return;
  const float* q = Qw + (size_t)i * AA;
  float s = 0.f;
#pragma unroll 4
  for (int k = 0; k < AA; ++k) s += q[k] * Wv[k];
  s += bv[0];
  g[i] = 1.f / (1.f + __expf(-s));
}

// ---------------------------------------------------------------------------
// Kernel 3: fused scores (WMMA) -> LDS stripe -> softmax -> gate -> out
// grid = B * N/16, 256 threads (8 waves). LDS holds 16 x 2048 fp32 scores
// (~128 KB; CDNA5 WGP has 320 KB). Wave w computes cols [w*256, w*256+256).
// ---------------------------------------------------------------------------
__global__ __launch_bounds__(256) void attn(const float* __restrict__ Qw,
                                            const float* __restrict__ Kw,
                                            const float* __restrict__ g,
                                            float* __restrict__ out) {
  __shared__ float sc[16 * LDSTRIDE];   // 131,328 bytes

  const int blk  = blockIdx.x;
  const int b    = blk >> 7;
  const int r0   = (blk & 127) << 4;
  const int wave = threadIdx.x >> 5;
  const int lane = threadIdx.x & 31;
  const int lm   = lane & 15;
  const int lh   = lane >> 4;

  // Preload all 32 A-fragments of the Q stripe (reused by all 16 col tiles).
  const float* qrow = Qw + (size_t)(b * NN + r0 + lm) * AA;
  v2f afrag[32];
#pragma unroll
  for (int kk = 0; kk < 32; ++kk)
    afrag[kk] = *(const v2f*)(qrow + 4 * kk + 2 * lh);

  for (int t = 0; t < 16; ++t) {
    const int c0 = wave * 256 + t * 16;
    const float* krow = Kw + (size_t)(b * NN + c0 + lm) * AA;
    v8f acc = {};
#pragma unroll
    for (int kk = 0; kk < 32; ++kk) {
      const v2f bfrag = *(const v2f*)(krow + 4 * kk + 2 * lh);  // Kw^T B-frag
      acc = __builtin_amdgcn_wmma_f32_16x16x4_f32(false, afrag[kk], false,
                                                  bfrag, (short)0, acc,
                                                  false, false);
    }
#pragma unroll
    for (int v = 0; v < 8; ++v)
      sc[(v + 8 * lh) * LDSTRIDE + c0 + lm] = acc[v];
  }
  __syncthreads();

  // Softmax + gating: each wave owns 2 of the 16 rows.
#pragma unroll
  for (int rr = 0; rr < 2; ++rr) {
    const int m   = wave * 2 + rr;
    const int row = r0 + m;                 // diagonal column index
    const float gv = g[b * NN + row];
    float* srow = sc + m * LDSTRIDE;

    float mx = -3.402823466e38f;
    for (int n = lane; n < NN; n += 32) {
      const float s = (n == row) ? -3.402823466e38f : srow[n];
      mx = fmaxf(mx, s);
    }
#pragma unroll
    for (int off = 16; off > 0; off >>= 1)
      mx = fmaxf(mx, __shfl_xor(mx, off, 32));

    float sum = 0.f;
    for (int n = lane; n < NN; n += 32) {
      const float e = (n == row) ? 0.f : __expf(srow[n] - mx);
      srow[n] = e;
      sum += e;
    }
#pragma unroll
    for (int off = 16; off > 0; off >>= 1)
      sum += __shfl_xor(sum, off, 32);

    const float scale = (1.f - gv) / sum;
    float* orow = out + (size_t)(b * NN + row) * NN;
    for (int n = lane; n < NN; n += 32)
      orow[n] = (n == row) ? gv : srow[n] * scale;
  }
}

// ---------------------------------------------------------------------------
extern "C" void kernel_launch(void* const* d_in, const int* in_sizes, int n_in,
                              void* d_out, int out_size, void* d_ws,
                              size_t ws_size, hipStream_t stream) {
  const float* x  = (const float*)d_in[0];
  const float* Wq = (const float*)d_in[1];
  const float* bq = (const float*)d_in[2];
  const float* Wk = (const float*)d_in[3];
  const float* bk = (const float*)d_in[4];
  const float* Wv = (const float*)d_in[5];
  const float* bv = (const float*)d_in[6];
  float* out = (float*)d_out;

  float* Qw = (float*)d_ws;                       // 8 MB
  float* Kw = Qw + (size_t)BB * NN * AA;          // 8 MB
  float* gg = Kw + (size_t)BB * NN * AA;          // 64 KB

  proj_qk<<<BB * (NN / 16), 256, 0, stream>>>(x, Wq, bq, Wk, bk, Qw, Kw);
  gate_kernel<<<(BB * NN + 255) / 256, 256, 0, stream>>>(Qw, Wv, bv, gg);
  attn<<<BB * (NN / 16), 256, 0, stream>>>(Qw, Kw, gg, out);
}